// IsotonicLayer_28956669510291
// MI455X (gfx1250) — compile-verified
//
#include <hip/hip_runtime.h>

// Isotonic layer: out = sigmoid(STEP*csum[u,idx] + delta*w[u,idx] + RESIDUE + b[u])
//   x: (65536, 256) f32, v: (256, 501) f32, b: (256,) f32 -> out: (65536, 256) f32
//
// Memory-bound: 64 MB in + 64 MB out -> ~5.5 us floor at 23.3 TB/s.
// Strategy: precompute per-unit interleaved table (A = STEP*excl_cumsum + RESIDUE + b, w)
// in d_ws (1 MB), stage 32 units' rows (128 KB) into LDS per block via CDNA5
// async global->LDS loads, gather 8B/element with ds_load_b64, fma + sigmoid.

#define UNITS 256
#define NB 501           // num buckets
#define UG 32            // units per block (32 * 501 * 8B = 128,256 B LDS)
#define ROWS_PER_BLOCK 1024
#define BATCH 65536

#if defined(__gfx1250__) && __has_builtin(__builtin_amdgcn_global_load_async_to_lds_b128) && __has_builtin(__builtin_amdgcn_s_wait_asynccnt)
#define USE_ASYNC_LDS 1
#else
#define USE_ASYNC_LDS 0
#endif

typedef int v4i __attribute__((vector_size(16)));  // matches builtin's int4 pointee

// ---------------- prep: relu + exclusive scan over buckets, fold in b/RESIDUE ----------
__global__ __launch_bounds__(512) void iso_prep(const float* __restrict__ v,
                                                const float* __restrict__ b,
                                                float2* __restrict__ tab) {
  __shared__ float s[512];
  const int u = blockIdx.x;   // one unit per block (256 blocks)
  const int t = threadIdx.x;  // 512 threads cover 501 buckets

  float w = 0.0f;
  if (t < NB) {
    float vv = v[u * NB + t];
    w = vv > 0.0f ? vv : 0.0f;  // relu(v)
  }
  s[t] = w;
  __syncthreads();

  // inclusive Hillis-Steele scan over 512 slots (log2 = 9 steps)
#pragma unroll
  for (int off = 1; off < 512; off <<= 1) {
    float cur = s[t];
    float add = (t >= off) ? s[t - off] : 0.0f;
    __syncthreads();
    s[t] = cur + add;
    __syncthreads();
  }

  if (t < NB) {
    float excl = (t == 0) ? 0.0f : s[t - 1];    // exact exclusive cumsum
    float A = 0.05f * excl + (-17.05f + b[u]);  // STEP*csum + RESIDUE + b
    tab[u * NB + t] = make_float2(A, w);
  }
}

// ---------------- main: bucketize + LDS gather + sigmoid --------------------------------
__device__ __forceinline__ float iso_eval(float xi, const float2* __restrict__ row) {
  float xc = fminf(fmaxf(xi, -17.0f), 8.0f);     // clip (LB+1e-9, UB-1e-9 round to +-bounds in f32)
  float xp = (xc + 17.0f) + 0.05f;               // x_clipped - LB + STEP
  int idx = (int)(xp * 20.0f);                   // / STEP  (logits continuous across bucket edges)
  idx = idx < NB - 1 ? idx : NB - 1;
  float delta = fmaf((float)idx, -0.05f, xp);    // xp - idx*STEP
  float2 ab = row[idx];                          // ds_load_b64 gather
  float logit = fmaf(delta, ab.y, ab.x);         // A + delta*w
  return 1.0f / (1.0f + __expf(-logit));
}

__global__ __launch_bounds__(256) void iso_main(const float* __restrict__ x,
                                                const float2* __restrict__ tab,
                                                float* __restrict__ out) {
  __shared__ float2 s_tab[UG * NB];  // 128,256 bytes (gfx1250: 320 KB/WGP -> 2 blocks/WGP)
  const int t = threadIdx.x;
  const int unitBase = blockIdx.y * UG;

  // Stage this block's 32 table rows (contiguous 128 KB in global) into LDS.
  {
    const float4* __restrict__ src = (const float4*)(tab + (size_t)unitBase * NB);
    float4* dst = (float4*)s_tab;
    const int chunks = (UG * NB) / 2;  // 8016 x 16B
#if USE_ASYNC_LDS
    for (int i = t; i < chunks; i += 256) {
      __builtin_amdgcn_global_load_async_to_lds_b128(
          (__attribute__((address_space(1))) v4i*)(src + i),
          (__attribute__((address_space(3))) v4i*)(dst + i),
          0, 0);
    }
    __builtin_amdgcn_s_wait_asynccnt(0);
#else
    for (int i = t; i < chunks; i += 256) dst[i] = src[i];
#endif
  }
  __syncthreads();

  // Thread layout: 8 unit-quads x 32 rows per pass; 8 consecutive threads read
  // 128 contiguous bytes of x (full cacheline), float4 loads/stores.
  const int q = t & 7;    // unit quad (4 units each)
  const int rs = t >> 3;  // row sub-index 0..31
  const float2* __restrict__ r0 = s_tab + (q * 4 + 0) * NB;
  const float2* __restrict__ r1 = s_tab + (q * 4 + 1) * NB;
  const float2* __restrict__ r2 = s_tab + (q * 4 + 2) * NB;
  const float2* __restrict__ r3 = s_tab + (q * 4 + 3) * NB;

  const size_t rowBase = (size_t)blockIdx.x * ROWS_PER_BLOCK + rs;
  const size_t colOff = (size_t)unitBase + q * 4;

#pragma unroll 4
  for (int it = 0; it < ROWS_PER_BLOCK / 32; ++it) {
    const size_t base = (rowBase + (size_t)it * 32) * UNITS + colOff;
    const float4 xv = *(const float4*)(x + base);
    float4 ov;
    ov.x = iso_eval(xv.x, r0);
    ov.y = iso_eval(xv.y, r1);
    ov.z = iso_eval(xv.z, r2);
    ov.w = iso_eval(xv.w, r3);
    *(float4*)(out + base) = ov;
  }
}

extern "C" void kernel_launch(void* const* d_in, const int* in_sizes, int n_in,
                              void* d_out, int out_size, void* d_ws, size_t ws_size,
                              hipStream_t stream) {
  const float* x = (const float*)d_in[0];
  const float* v = (const float*)d_in[1];
  const float* b = (const float*)d_in[2];
  float* out = (float*)d_out;
  float2* tab = (float2*)d_ws;  // 256*501*8 = 1,026,048 bytes of scratch

  iso_prep<<<UNITS, 512, 0, stream>>>(v, b, tab);

  dim3 grid(BATCH / ROWS_PER_BLOCK, UNITS / UG);  // (64, 8)
  iso_main<<<grid, 256, 0, stream>>>(x, tab, out);
}